// SSFE_27530740367931
// MI455X (gfx1250) — compile-verified
//
#include <hip/hip_runtime.h>
#include <hip/hip_bf16.h>
#include <math.h>

typedef __attribute__((ext_vector_type(2))) float v2f;
typedef __attribute__((ext_vector_type(8))) float v8f;

#define NB   16      // batch
#define CT   896     // total channels (128+256+384..512)
#define HW   4096    // 64*64
#define NC   (NB*CT) // 14336 (b,c) pairs
#define EPSV 1e-5f

__device__ __forceinline__ float gelu_exact(float x) {
    return 0.5f * x * (1.0f + erff(x * 0.70710678118654752f));
}
__device__ __forceinline__ float sigmoidf(float x) {
    return 1.0f / (1.0f + expf(-x));
}

// ---------------------------------------------------------------------------
// Pass 1: per-(b,c) mean, mean-of-squares, max, min of the raw inputs.
// One 256-thread block per contiguous 4096-float channel slice.
// ---------------------------------------------------------------------------
__global__ void k_stats(const float* __restrict__ d0, const float* __restrict__ d1,
                        const float* __restrict__ d2,
                        float* __restrict__ mean, float* __restrict__ msq,
                        float* __restrict__ mx, float* __restrict__ mn) {
    int bid = blockIdx.x;            // 0..NC-1
    int b = bid / CT, c = bid % CT;
    const float* src;
    if (c < 128)      src = d0 + ((size_t)b * 128 + c)        * HW;
    else if (c < 384) src = d1 + ((size_t)b * 256 + (c - 128)) * HW;
    else              src = d2 + ((size_t)b * 512 + (c - 384)) * HW;
    const float4* p4 = (const float4*)src;
    int tid = threadIdx.x;
    float s = 0.f, sq = 0.f, hi = -3.402823466e38f, lo = 3.402823466e38f;
    for (int i = tid; i < HW / 4; i += 256) {
        float4 v = p4[i];
        s  += (v.x + v.y) + (v.z + v.w);
        sq += (v.x * v.x + v.y * v.y) + (v.z * v.z + v.w * v.w);
        hi = fmaxf(hi, fmaxf(fmaxf(v.x, v.y), fmaxf(v.z, v.w)));
        lo = fminf(lo, fminf(fminf(v.x, v.y), fminf(v.z, v.w)));
    }
    __shared__ float rs[256], rq[256], rh[256], rl[256];
    rs[tid] = s; rq[tid] = sq; rh[tid] = hi; rl[tid] = lo;
    __syncthreads();
    for (int off = 128; off > 0; off >>= 1) {
        if (tid < off) {
            rs[tid] += rs[tid + off];
            rq[tid] += rq[tid + off];
            rh[tid] = fmaxf(rh[tid], rh[tid + off]);
            rl[tid] = fminf(rl[tid], rl[tid + off]);
        }
        __syncthreads();
    }
    if (tid == 0) {
        mean[bid] = rs[0] * (1.f / HW);
        msq[bid]  = rq[0] * (1.f / HW);
        mx[bid]   = rh[0];
        mn[bid]   = rl[0];
    }
}

// ---------------------------------------------------------------------------
// Branch GroupNorm stats (analytic): a0pre = rs1*gn_w, b0pre = gn_b - mu1*a0pre
// gate input p[b,c] = a0pre*mean + b0pre. One block per batch.
// ---------------------------------------------------------------------------
__global__ void k_prep(const float* __restrict__ mean, const float* __restrict__ msq,
                       const float* __restrict__ gn0w, const float* __restrict__ gn0b,
                       const float* __restrict__ gn1w, const float* __restrict__ gn1b,
                       const float* __restrict__ gn2w, const float* __restrict__ gn2b,
                       float* __restrict__ a0pre, float* __restrict__ b0pre,
                       float* __restrict__ p) {
    int b = blockIdx.x, t = threadIdx.x;
    if (t >= 96) return;
    int br = t >> 5, g = t & 31;
    const int coffs[3] = {0, 128, 384};
    const int cpgs[3]  = {4, 8, 16};
    const float* gw = (br == 0) ? gn0w : (br == 1) ? gn1w : gn2w;
    const float* gb = (br == 0) ? gn0b : (br == 1) ? gn1b : gn2b;
    int cpg = cpgs[br];
    int lbase = g * cpg;
    int base = b * CT + coffs[br] + lbase;
    float sm = 0.f, sq = 0.f;
    for (int j = 0; j < cpg; j++) { sm += mean[base + j]; sq += msq[base + j]; }
    float inv = 1.0f / (float)cpg;
    float mu = sm * inv;
    float var = sq * inv - mu * mu;
    float rs = rsqrtf(var + EPSV);
    for (int j = 0; j < cpg; j++) {
        float a = rs * gw[lbase + j];
        float bb = gb[lbase + j] - mu * a;
        a0pre[base + j] = a;
        b0pre[base + j] = bb;
        p[base + j] = a * mean[base + j] + bb;
    }
}

// ---------------------------------------------------------------------------
// Single-wave fp32 WMMA 16x16 tile:  D[m][n] = sum_k A[m*lda+k] * Bw[n*ldb+k]
// V_WMMA_F32_16X16X4_F32: A 16x4 (lanes 0-15: K0,K1; 16-31: K2,K3),
// B 4x16 mirrored, C/D: lanes 0-15 rows 0-7, lanes 16-31 rows 8-15.
// Requires blockDim.x == 32, no divergence (EXEC all-1s).
// ---------------------------------------------------------------------------
__device__ __forceinline__ v8f wmma_tile_f32(const float* __restrict__ A, int lda,
                                             const float* __restrict__ Bw, int ldb,
                                             int K, int ncol, int lane) {
    int m  = lane & 15;
    int kk = (lane >> 4) << 1;          // 0 for lanes 0-15, 2 for lanes 16-31
    const float* ap = A + m * lda + kk;
    const float* bp = Bw + ncol * ldb + kk;
    v8f acc = {0.f, 0.f, 0.f, 0.f, 0.f, 0.f, 0.f, 0.f};
    for (int k0 = 0; k0 < K; k0 += 4) {
        v2f av; av.x = ap[k0]; av.y = ap[k0 + 1];
        v2f bv; bv.x = bp[k0]; bv.y = bp[k0 + 1];
        acc = __builtin_amdgcn_wmma_f32_16x16x4_f32(false, av, false, bv,
                                                    (short)0, acc, false, false);
    }
    return acc;
}

// Gate MLP stage 1: h = gelu(p @ w1^T + b1) for all 3 branches. 14 tiles.
__global__ void k_gate1(const float* __restrict__ p,
                        const float* __restrict__ w1_0, const float* __restrict__ b1_0,
                        const float* __restrict__ w1_1, const float* __restrict__ b1_1,
                        const float* __restrict__ w1_2, const float* __restrict__ b1_2,
                        float* __restrict__ hb) {
    int bid = blockIdx.x, lane = threadIdx.x;
    int br, nt;
    if (bid < 2)      { br = 0; nt = bid; }
    else if (bid < 6) { br = 1; nt = bid - 2; }
    else              { br = 2; nt = bid - 6; }
    const int coffs[3] = {0, 128, 384}, Cs[3] = {128, 256, 512}, hoffs[3] = {0, 32, 96};
    const float* w1 = (br == 0) ? w1_0 : (br == 1) ? w1_1 : w1_2;
    const float* b1 = (br == 0) ? b1_0 : (br == 1) ? b1_1 : b1_2;
    int Kd = Cs[br];
    int ncol = nt * 16 + (lane & 15);
    v8f acc = wmma_tile_f32(p + coffs[br], CT, w1, Kd, Kd, ncol, lane);
    float bias = b1[ncol];
    int rbase = (lane >> 4) << 3;
    for (int r = 0; r < 8; r++) {
        int row = rbase + r;  // batch index
        hb[row * 224 + hoffs[br] + ncol] = gelu_exact(acc[r] + bias);
    }
}

// Gate MLP stage 2: g = sigmoid(h @ w2^T + b2). 56 tiles.
__global__ void k_gate2(const float* __restrict__ hb,
                        const float* __restrict__ w2_0, const float* __restrict__ b2_0,
                        const float* __restrict__ w2_1, const float* __restrict__ b2_1,
                        const float* __restrict__ w2_2, const float* __restrict__ b2_2,
                        float* __restrict__ gbuf) {
    int bid = blockIdx.x, lane = threadIdx.x;
    int br, nt;
    if (bid < 8)       { br = 0; nt = bid; }
    else if (bid < 24) { br = 1; nt = bid - 8; }
    else               { br = 2; nt = bid - 24; }
    const int coffs[3] = {0, 128, 384}, hids[3] = {32, 64, 128}, hoffs[3] = {0, 32, 96};
    const float* w2 = (br == 0) ? w2_0 : (br == 1) ? w2_1 : w2_2;
    const float* b2 = (br == 0) ? b2_0 : (br == 1) ? b2_1 : b2_2;
    int hid = hids[br];
    int ncol = nt * 16 + (lane & 15);
    v8f acc = wmma_tile_f32(hb + hoffs[br], 224, w2, hid, hid, ncol, lane);
    float bias = b2[ncol];
    int rbase = (lane >> 4) << 3;
    for (int r = 0; r < 8; r++) {
        int row = rbase + r;
        gbuf[row * CT + coffs[br] + ncol] = sigmoidf(acc[r] + bias);
    }
}

// ---------------------------------------------------------------------------
// Second GroupNorm (analytic over gated affine channels) + gp/mp + router.
// One block per batch.
// ---------------------------------------------------------------------------
__global__ void k_gn2(const float* __restrict__ mean, const float* __restrict__ msq,
                      const float* __restrict__ mx, const float* __restrict__ mn,
                      const float* __restrict__ a0pre, const float* __restrict__ b0pre,
                      const float* __restrict__ gbuf,
                      const float* __restrict__ cnw, const float* __restrict__ cnb,
                      const float* __restrict__ rw, const float* __restrict__ rb,
                      float* __restrict__ A1, float* __restrict__ B1,
                      float* __restrict__ gp, float* __restrict__ mp,
                      float* __restrict__ alpha, float* __restrict__ beta) {
    int b = blockIdx.x, t = threadIdx.x;
    __shared__ float sEy[CT], sEy2[CT], sA0[CT], sB0[CT], sGP[CT];
    __shared__ float sMu[32], sRs[32];
    __shared__ float red0[256], red1[256];
    for (int c = t; c < CT; c += 256) {
        int idx = b * CT + c;
        float g  = gbuf[idx];
        float A0 = g * a0pre[idx];
        float B0 = g * b0pre[idx];
        float m = mean[idx], q = msq[idx];
        sA0[c] = A0; sB0[c] = B0;
        sEy[c]  = A0 * m + B0;
        sEy2[c] = A0 * A0 * q + 2.f * A0 * B0 * m + B0 * B0;
    }
    __syncthreads();
    if (t < 32) {
        float sm = 0.f, sq = 0.f;
        for (int j = 0; j < 28; j++) { sm += sEy[t * 28 + j]; sq += sEy2[t * 28 + j]; }
        float mu = sm * (1.f / 28.f);
        float var = sq * (1.f / 28.f) - mu * mu;
        sMu[t] = mu;
        sRs[t] = rsqrtf(var + EPSV);
    }
    __syncthreads();
    for (int c = t; c < CT; c += 256) {
        int G = c / 28;
        int idx = b * CT + c;
        float rs2 = sRs[G], mu2 = sMu[G];
        float a1  = sA0[c] * rs2 * cnw[c];
        float b1v = (sB0[c] - mu2) * rs2 * cnw[c] + cnb[c];
        A1[idx] = a1; B1[idx] = b1v;
        float gpv = a1 * mean[idx] + b1v;
        float mpv = ((a1 >= 0.f) ? a1 * mx[idx] : a1 * mn[idx]) + b1v;
        gp[idx] = gpv; mp[idx] = mpv;
        sGP[c] = gpv;
    }
    __syncthreads();
    float s0 = 0.f, s1 = 0.f;
    for (int c = t; c < CT; c += 256) { s0 += sGP[c] * rw[c]; s1 += sGP[c] * rw[CT + c]; }
    red0[t] = s0; red1[t] = s1;
    __syncthreads();
    for (int off = 128; off > 0; off >>= 1) {
        if (t < off) { red0[t] += red0[t + off]; red1[t] += red1[t + off]; }
        __syncthreads();
    }
    if (t == 0) {
        float l0 = red0[0] + rb[0], l1 = red1[0] + rb[1];
        float mm = fmaxf(l0, l1);
        float e0 = expf(l0 - mm), e1 = expf(l1 - mm);
        float inv = 1.f / (e0 + e1);
        alpha[b] = e0 * inv;
        beta[b]  = e1 * inv;
    }
}

// SE MLP stage 1 (ge on gp, le on mp): h = gelu(x @ w1^T + b1). 14 tiles.
__global__ void k_se1(const float* __restrict__ gp, const float* __restrict__ mp,
                      const float* __restrict__ gew1, const float* __restrict__ geb1,
                      const float* __restrict__ lew1, const float* __restrict__ leb1,
                      float* __restrict__ hse) {
    int bid = blockIdx.x, lane = threadIdx.x;
    int which = bid / 7, nt = bid % 7;
    const float* A  = which ? mp : gp;
    const float* w1 = which ? lew1 : gew1;
    const float* b1 = which ? leb1 : geb1;
    int ncol = nt * 16 + (lane & 15);
    v8f acc = wmma_tile_f32(A, CT, w1, CT, CT, ncol, lane);
    float bias = b1[ncol];
    int rbase = (lane >> 4) << 3;
    for (int r = 0; r < 8; r++) {
        int row = rbase + r;
        hse[which * (NB * 112) + row * 112 + ncol] = gelu_exact(acc[r] + bias);
    }
}

// SE MLP stage 2: w = sigmoid((h @ w2^T + b2)/t). 112 tiles.
__global__ void k_se2(const float* __restrict__ hse,
                      const float* __restrict__ gew2, const float* __restrict__ geb2,
                      const float* __restrict__ lew2, const float* __restrict__ leb2,
                      const float* __restrict__ temp,
                      float* __restrict__ wg, float* __restrict__ wl) {
    int bid = blockIdx.x, lane = threadIdx.x;
    int which = bid / 56, nt = bid % 56;
    const float* w2 = which ? lew2 : gew2;
    const float* b2 = which ? leb2 : geb2;
    float tcl = fminf(fmaxf(temp[0], 0.5f), 2.0f);
    float ti = 1.0f / tcl;
    int ncol = nt * 16 + (lane & 15);
    v8f acc = wmma_tile_f32(hse + which * (NB * 112), 112, w2, 112, 112, ncol, lane);
    float bias = b2[ncol];
    float* dst = which ? wl : wg;
    int rbase = (lane >> 4) << 3;
    for (int r = 0; r < 8; r++) {
        int row = rbase + r;
        dst[row * CT + ncol] = sigmoidf((acc[r] + bias) * ti);
    }
}

// ---------------------------------------------------------------------------
// Pass 2: out = S*raw + T  with S = A1*(w_ch+gamma), T = B1*(w_ch+gamma).
// ---------------------------------------------------------------------------
__global__ void k_out(const float* __restrict__ d0, const float* __restrict__ d1,
                      const float* __restrict__ d2,
                      const float* __restrict__ A1, const float* __restrict__ B1,
                      const float* __restrict__ wg, const float* __restrict__ wl,
                      const float* __restrict__ alpha, const float* __restrict__ beta,
                      const float* __restrict__ gamma,
                      float* __restrict__ out) {
    int bid = blockIdx.x;
    int b = bid / CT, c = bid % CT;
    __shared__ float sS, sT;
    if (threadIdx.x == 0) {
        float wgv = wg[bid], wlv = wl[bid];
        float wch = alpha[b] * wgv + beta[b] * wlv * wgv;
        float f = wch + gamma[0];
        sS = A1[bid] * f;
        sT = B1[bid] * f;
    }
    __syncthreads();
    float S = sS, T = sT;
    const float* src;
    if (c < 128)      src = d0 + ((size_t)b * 128 + c)        * HW;
    else if (c < 384) src = d1 + ((size_t)b * 256 + (c - 128)) * HW;
    else              src = d2 + ((size_t)b * 512 + (c - 384)) * HW;
    const float4* p4 = (const float4*)src;
    float4* o4 = (float4*)(out + (size_t)bid * HW);
    for (int i = threadIdx.x; i < HW / 4; i += 256) {
        float4 v = p4[i];
        float4 r;
        r.x = fmaf(S, v.x, T);
        r.y = fmaf(S, v.y, T);
        r.z = fmaf(S, v.z, T);
        r.w = fmaf(S, v.w, T);
        o4[i] = r;
    }
}

extern "C" void kernel_launch(void* const* d_in, const int* in_sizes, int n_in,
                              void* d_out, int out_size, void* d_ws, size_t ws_size,
                              hipStream_t stream) {
    // setup_inputs() dict order
    const float* d0    = (const float*)d_in[0];
    const float* d1    = (const float*)d_in[1];
    const float* d2    = (const float*)d_in[2];
    const float* gn0w  = (const float*)d_in[3];
    const float* gn0b  = (const float*)d_in[4];
    const float* g0w1  = (const float*)d_in[5];
    const float* g0b1  = (const float*)d_in[6];
    const float* g0w2  = (const float*)d_in[7];
    const float* g0b2  = (const float*)d_in[8];
    const float* gn1w  = (const float*)d_in[9];
    const float* gn1b  = (const float*)d_in[10];
    const float* g1w1  = (const float*)d_in[11];
    const float* g1b1  = (const float*)d_in[12];
    const float* g1w2  = (const float*)d_in[13];
    const float* g1b2  = (const float*)d_in[14];
    const float* gn2w  = (const float*)d_in[15];
    const float* gn2b  = (const float*)d_in[16];
    const float* g2w1  = (const float*)d_in[17];
    const float* g2b1  = (const float*)d_in[18];
    const float* g2w2  = (const float*)d_in[19];
    const float* g2b2  = (const float*)d_in[20];
    const float* cnw   = (const float*)d_in[21];
    const float* cnb   = (const float*)d_in[22];
    const float* gew1  = (const float*)d_in[23];
    const float* geb1  = (const float*)d_in[24];
    const float* gew2  = (const float*)d_in[25];
    const float* geb2  = (const float*)d_in[26];
    const float* lew1  = (const float*)d_in[27];
    const float* leb1  = (const float*)d_in[28];
    const float* lew2  = (const float*)d_in[29];
    const float* leb2  = (const float*)d_in[30];
    const float* rw    = (const float*)d_in[31];
    const float* rb    = (const float*)d_in[32];
    const float* temp  = (const float*)d_in[33];
    const float* gamma = (const float*)d_in[34];
    float* out = (float*)d_out;

    // workspace layout (floats)
    float* W      = (float*)d_ws;
    float* w_mean = W + 0 * NC;
    float* w_msq  = W + 1 * NC;
    float* w_max  = W + 2 * NC;
    float* w_min  = W + 3 * NC;
    float* w_a0   = W + 4 * NC;
    float* w_b0   = W + 5 * NC;
    float* w_p    = W + 6 * NC;
    float* w_g    = W + 7 * NC;
    float* w_A1   = W + 8 * NC;
    float* w_B1   = W + 9 * NC;
    float* w_gp   = W + 10 * NC;
    float* w_mp   = W + 11 * NC;
    float* w_wg   = W + 12 * NC;
    float* w_wl   = W + 13 * NC;
    float* w_hb   = W + 14 * NC;              // 16*224
    float* w_hse  = W + 14 * NC + NB * 224;   // 2*16*112
    float* w_al   = W + 14 * NC + NB * 224 + 2 * NB * 112;
    float* w_be   = w_al + NB;

    k_stats<<<NC, 256, 0, stream>>>(d0, d1, d2, w_mean, w_msq, w_max, w_min);
    k_prep<<<NB, 128, 0, stream>>>(w_mean, w_msq, gn0w, gn0b, gn1w, gn1b, gn2w, gn2b,
                                   w_a0, w_b0, w_p);
    k_gate1<<<14, 32, 0, stream>>>(w_p, g0w1, g0b1, g1w1, g1b1, g2w1, g2b1, w_hb);
    k_gate2<<<56, 32, 0, stream>>>(w_hb, g0w2, g0b2, g1w2, g1b2, g2w2, g2b2, w_g);
    k_gn2<<<NB, 256, 0, stream>>>(w_mean, w_msq, w_max, w_min, w_a0, w_b0, w_g,
                                  cnw, cnb, rw, rb, w_A1, w_B1, w_gp, w_mp, w_al, w_be);
    k_se1<<<14, 32, 0, stream>>>(w_gp, w_mp, gew1, geb1, lew1, leb1, w_hse);
    k_se2<<<112, 32, 0, stream>>>(w_hse, gew2, geb2, lew2, leb2, temp, w_wg, w_wl);
    k_out<<<NC, 256, 0, stream>>>(d0, d1, d2, w_A1, w_B1, w_wg, w_wl, w_al, w_be,
                                  gamma, out);
    (void)in_sizes; (void)n_in; (void)out_size; (void)ws_size;
}